// StateProbabilities_22024592293959
// MI455X (gfx1250) — compile-verified
//
#include <hip/hip_runtime.h>
#include <hip/hip_bf16.h>

// CDNA5 (gfx1250) WMMA types
typedef float v2f __attribute__((ext_vector_type(2)));
typedef float v8f __attribute__((ext_vector_type(8)));

#define NDRIVES 16
#define PREN    64          // DIM*DIM
#define MAT     16          // embedded real matrix dimension (2*DIM)
#define TSTRIDE 17          // padded LDS row stride (avoids 16-float bank conflicts)
#define WAVES_PER_BLOCK 8
#define BATCH_PER_WAVE  16
#define TAYLOR_DEG 8

// D = A(16x4,f32) * B(4x16,f32) + C(16x16,f32)   -> v_wmma_f32_16x16x4_f32
__device__ __forceinline__ v8f wmma4_f32(v2f a, v2f b, v8f c) {
    return __builtin_amdgcn_wmma_f32_16x16x4_f32(
        /*neg_a=*/false, a, /*neg_b=*/false, b,
        /*c_mod=*/(short)0, c, /*reuse_a=*/false, /*reuse_b=*/false);
}

__global__ __launch_bounds__(256)
void StateProbabilities_22024592293959_kernel(const float* __restrict__ x,
                                              const float* __restrict__ kern,
                                              const float* __restrict__ bias,
                                              float* __restrict__ out)
{
    __shared__ float preLds[WAVES_PER_BLOCK][BATCH_PER_WAVE][PREN];
    __shared__ float tLds[WAVES_PER_BLOCK][MAT][TSTRIDE];

    const int tid  = threadIdx.x;
    const int wave = tid >> 5;
    const int lane = tid & 31;
    const int lo   = lane & 15;   // row / col index within 16
    const int hi   = lane >> 4;   // half select

    float (*pre)[PREN]   = preLds[wave];
    float (*T)[TSTRIDE]  = tLds[wave];

    const int batchBase = (blockIdx.x * WAVES_PER_BLOCK + wave) * BATCH_PER_WAVE;

    // ---------------- Stage 1: pre = x_tile(16x16) @ kernel(16x64) + bias ----------------
    // A-layout (16x4 chunk c, vgpr v): element (row=lo, k=4c+2hi+v)
    v2f a[4];
    #pragma unroll
    for (int c = 0; c < 4; ++c)
        #pragma unroll
        for (int v = 0; v < 2; ++v)
            a[c][v] = x[(size_t)(batchBase + lo) * NDRIVES + (4*c + 2*hi + v)];

    #pragma unroll
    for (int nt = 0; nt < 4; ++nt) {              // 4 tiles of 16 output columns
        const float bcol = bias[nt*16 + lo];
        v8f acc;
        #pragma unroll
        for (int g = 0; g < 8; ++g) acc[g] = bcol;  // C preloaded with bias (col-only)
        #pragma unroll
        for (int c = 0; c < 4; ++c) {
            v2f bk;                                  // B-layout: (k=4c+2hi+v, n=lo)
            #pragma unroll
            for (int v = 0; v < 2; ++v)
                bk[v] = kern[(size_t)(4*c + 2*hi + v) * PREN + nt*16 + lo];
            acc = wmma4_f32(a[c], bk, acc);
        }
        #pragma unroll
        for (int g = 0; g < 8; ++g)                  // D-layout: (m=g+8hi, n=lo)
            pre[g + 8*hi][nt*16 + lo] = acc[g];
    }

    // ---------------- Stage 2: per batch element, exp(-iH) e0 via WMMA scaling&squaring --
    for (int b = 0; b < BATCH_PER_WAVE; ++b) {
        const float* pb = pre[b];

        // Build E = [[A, S],[-S, A]] (real embedding of -iH) in WMMA-A layout.
        v2f Ae[4];
        float fro2 = 0.f;
        #pragma unroll
        for (int c = 0; c < 4; ++c)
            #pragma unroll
            for (int v = 0; v < 2; ++v) {
                const int r  = lo;
                const int cl = 4*c + 2*hi + v;
                const int rr = r & 7, cc = cl & 7;
                const float prc = pb[rr*8 + cc];
                const float pcr = pb[cc*8 + rr];
                const float anti = prc - pcr;        // Re(-iH) = pre - pre^T
                const float symm = prc + pcr;        // -Im(-iH) block
                const float e = (r < 8) ? ((cl < 8) ?  anti : symm)
                                        : ((cl < 8) ? -symm : anti);
                Ae[c][v] = e;
                fro2 += e * e;
            }
        // wave-wide Frobenius reduction (A-layout covers each element exactly once)
        #pragma unroll
        for (int m = 16; m >= 1; m >>= 1)
            fro2 += __shfl_xor(fro2, m, 32);
        const float fro = sqrtf(fro2);
        int s = 0;
        if (fro > 0.5f) s = (int)ceilf(log2f(fro * 2.0f));   // theta = 0.5
        if (s < 0)  s = 0;
        if (s > 24) s = 24;
        s = __builtin_amdgcn_readfirstlane(s);               // wave-uniform (EXEC stays full)
        const float scale = exp2f((float)(-s));
        #pragma unroll
        for (int c = 0; c < 4; ++c) { Ae[c][0] *= scale; Ae[c][1] *= scale; }

        // T = I
        #pragma unroll
        for (int g = 0; g < 8; ++g)
            T[g + 8*hi][lo] = (g + 8*hi == lo) ? 1.f : 0.f;

        // Horner Taylor: T <- I + (E/m) * T, m = DEG..1  (E stays in registers)
        #pragma unroll
        for (int m = TAYLOR_DEG; m >= 1; --m) {
            v8f d = {};
            #pragma unroll
            for (int c = 0; c < 4; ++c) {
                v2f bk;
                bk[0] = T[4*c + 2*hi + 0][lo];
                bk[1] = T[4*c + 2*hi + 1][lo];
                d = wmma4_f32(Ae[c], bk, d);
            }
            const float inv = 1.0f / (float)m;
            #pragma unroll
            for (int g = 0; g < 8; ++g) {
                const int row = g + 8*hi;
                T[row][lo] = d[g] * inv + ((row == lo) ? 1.f : 0.f);
            }
        }

        // s squarings: T <- T*T  (D->A/B relayout through padded LDS; same-wave DS is in-order)
        for (int i = 0; i < s; ++i) {
            v2f ta[4], tb[4];
            #pragma unroll
            for (int c = 0; c < 4; ++c) {
                ta[c][0] = T[lo][4*c + 2*hi + 0];
                ta[c][1] = T[lo][4*c + 2*hi + 1];
                tb[c][0] = T[4*c + 2*hi + 0][lo];
                tb[c][1] = T[4*c + 2*hi + 1][lo];
            }
            v8f d = {};
            #pragma unroll
            for (int c = 0; c < 4; ++c)
                d = wmma4_f32(ta[c], tb[c], d);
            #pragma unroll
            for (int g = 0; g < 8; ++g)
                T[g + 8*hi][lo] = d[g];
        }

        // out[b, j] = |U[j,0]|^2 = expE[j,0]^2 + expE[j+8,0]^2
        if (lane < 8) {
            const float re = T[lane][0];
            const float im = T[lane + 8][0];
            out[(size_t)(batchBase + b) * 8 + lane] = re*re + im*im;
        }
    }
}

extern "C" void kernel_launch(void* const* d_in, const int* in_sizes, int n_in,
                              void* d_out, int out_size, void* d_ws, size_t ws_size,
                              hipStream_t stream) {
    const float* x    = (const float*)d_in[0];
    const float* kern = (const float*)d_in[1];
    const float* bias = (const float*)d_in[2];
    float* out = (float*)d_out;

    const int batch  = in_sizes[0] / NDRIVES;                      // 131072
    const int blocks = batch / (WAVES_PER_BLOCK * BATCH_PER_WAVE); // 1024

    StateProbabilities_22024592293959_kernel<<<dim3(blocks), dim3(256), 0, stream>>>(
        x, kern, bias, out);
}